// SpGATVAE_28200755265681
// MI455X (gfx1250) — compile-verified
//
#include <hip/hip_runtime.h>
#include <hip/hip_bf16.h>
#include <cstdint>

typedef float v2f __attribute__((ext_vector_type(2)));
typedef float v8f __attribute__((ext_vector_type(8)));

#define NFEAT  512
#define NHID   64
#define NHEADS 8
#define HCOLS  (NHEADS * NHID)   // 512
#define NOUT   128
#define ALPHA  0.2f
#define KC     64                // K-chunk staged in LDS
#define PSTRIDE 160              // dwords per K-pair row; 160 % 64 == 32 -> half-waves
                                 // hit disjoint bank halves on b64 reads

// ---- repack Ws [8][512][64] -> Wall [512][512] (col = head*64 + j) ----
__global__ void repack_W(const float* __restrict__ Ws, float* __restrict__ Wall) {
    int t = blockIdx.x * blockDim.x + threadIdx.x;       // t = k*512 + c
    if (t >= NFEAT * HCOLS) return;
    int k = t >> 9, c = t & 511;
    Wall[t] = Ws[(size_t)(c >> 6) * (NFEAT * NHID) + (size_t)k * NHID + (c & 63)];
}

// ---- fp32 GEMM: C[M,N] = A[M,K] * B[K,N] via V_WMMA_F32_16X16X4_F32 ----
// Block = 8 waves; each wave computes a 16x64 strip (4 accumulators, A-frag
// reused 4x). B panel (KCx64) staged in LDS with K-pair interleaving so each
// B fragment {B[k][n],B[k+1][n]} is one aligned ds_load_b64.
// Requires: K % KC == 0, N % 64 == 0.
__global__ void gemm_wmma_f32(const float* __restrict__ A, const float* __restrict__ B,
                              float* __restrict__ C, int M, int N, int K) {
    __shared__ __align__(16) float Bs[(KC / 2) * PSTRIDE];
    const int lane = threadIdx.x & 31;
    const int wave = threadIdx.x >> 5;
    const int m    = lane & 15;          // A row in tile / C+B col in tile
    const int hi   = lane >> 4;          // half-wave select
    const int kb   = hi * 2;             // K sub-offset of this half-wave

    const int tilesM  = (M + 15) >> 4;
    const int tileM   = blockIdx.x * 8 + wave;     // 16 rows per wave
    const int colBase = blockIdx.y * 64;           // 64 cols per block
    const bool active = tileM < tilesM;

    int row  = tileM * 16 + m;
    int rowc = (active && row < M) ? row : 0;      // clamp: keep EXEC full, stay in barriers
    const float* __restrict__ Arow = A + (size_t)rowc * K;

    v8f acc0 = {0.f,0.f,0.f,0.f,0.f,0.f,0.f,0.f};
    v8f acc1 = acc0, acc2 = acc0, acc3 = acc0;

    for (int kc = 0; kc < K; kc += KC) {
        // cooperative staging (coalesced b128 from global, K-pair interleave to LDS):
        // Bs[(r>>1)*PSTRIDE + 2*c + (r&1)] = B[kc+r][colBase+c]
        for (int i = threadIdx.x * 4; i < KC * 64; i += 256 * 4) {
            int r = i >> 6, c = i & 63;
            const float4 v = *(const float4*)(B + (size_t)(kc + r) * N + colBase + c);
            float* p = &Bs[(r >> 1) * PSTRIDE + (r & 1) + 2 * c];
            p[0] = v.x; p[2] = v.y; p[4] = v.z; p[6] = v.w;
        }
        __syncthreads();

        #pragma unroll 4
        for (int k0 = 0; k0 < KC; k0 += 4) {
            v2f av = *(const v2f*)(Arow + kc + k0 + kb);           // global b64
            const float* __restrict__ b = &Bs[((k0 >> 1) + hi) * PSTRIDE + 2 * m];
            v2f b0 = *(const v2f*)(b + 0);                          // ds_load_b64
            v2f b1 = *(const v2f*)(b + 32);                         // +128 B
            v2f b2 = *(const v2f*)(b + 64);                         // +256 B
            v2f b3 = *(const v2f*)(b + 96);                         // +384 B
            acc0 = __builtin_amdgcn_wmma_f32_16x16x4_f32(false, av, false, b0, (short)0, acc0, false, false);
            acc1 = __builtin_amdgcn_wmma_f32_16x16x4_f32(false, av, false, b1, (short)0, acc1, false, false);
            acc2 = __builtin_amdgcn_wmma_f32_16x16x4_f32(false, av, false, b2, (short)0, acc2, false, false);
            acc3 = __builtin_amdgcn_wmma_f32_16x16x4_f32(false, av, false, b3, (short)0, acc3, false, false);
        }
        __syncthreads();
    }

    if (!active) return;
    float* __restrict__ Cbase = C + (size_t)colBase + m;
    #pragma unroll
    for (int i = 0; i < 8; ++i) {
        int r = tileM * 16 + i + hi * 8;           // C/D layout: M = i + 8*hi
        if (r < M) {
            float* cr = Cbase + (size_t)r * N;
            cr[0]  = acc0[i];
            cr[16] = acc1[i];
            cr[32] = acc2[i];
            cr[48] = acc3[i];
        }
    }
}

// ---- layer1 per-node attention scores: s_src/s_dst per (node, head) ----
__global__ void scores1(const float* __restrict__ H, const float* __restrict__ As,
                        float* __restrict__ ssrc, float* __restrict__ sdst, int Nn) {
    int t = blockIdx.x * blockDim.x + threadIdx.x;
    if (t >= Nn * NHEADS) return;
    int n = t >> 3, hd = t & 7;
    const float* __restrict__ h = H + (size_t)n * HCOLS + hd * NHID;
    const float* __restrict__ a = As + hd * (2 * NHID);
    float s0 = 0.f, s1 = 0.f;
    #pragma unroll 8
    for (int j = 0; j < NHID; ++j) { float v = h[j]; s0 += v * a[j]; s1 += v * a[NHID + j]; }
    ssrc[t] = s0; sdst[t] = s1;
}

// ---- layer1 per-edge coefficients e = exp(-leakyrelu(.)) + denom scatter ----
__global__ void edge_e1(const int* __restrict__ src, const int* __restrict__ dst,
                        const float* __restrict__ ssrc, const float* __restrict__ sdst,
                        float* __restrict__ e1, float* __restrict__ den1, int E) {
    int t = blockIdx.x * blockDim.x + threadIdx.x;
    if (t >= E * NHEADS) return;
    int edge = t >> 3, hd = t & 7;
    int si = src[edge], di = dst[edge];
    float x = ssrc[si * NHEADS + hd] + sdst[di * NHEADS + hd];
    float l = x > 0.f ? x : ALPHA * x;
    float e = expf(-l);
    e1[t] = e;
    atomicAdd(&den1[si * NHEADS + hd], e);
}

// ---- layer1 message scatter: num[src, c] += e * H[dst, c] ----
__global__ void scatter1(const int* __restrict__ src, const int* __restrict__ dst,
                         const float* __restrict__ e1, const float* __restrict__ H,
                         float* __restrict__ num1, int E) {
    int t = blockIdx.x * blockDim.x + threadIdx.x;
    if (t >= E * HCOLS) return;
    int edge = t >> 9, c = t & 511, hd = c >> 6;
    int si = src[edge], di = dst[edge];
    atomicAdd(&num1[(size_t)si * HCOLS + c],
              e1[edge * NHEADS + hd] * H[(size_t)di * HCOLS + c]);
}

// ---- layer1 normalize + ELU (in place: num1 becomes x2) ----
__global__ void norm_elu(float* __restrict__ num1, const float* __restrict__ den1, int Nn) {
    int t = blockIdx.x * blockDim.x + threadIdx.x;
    if (t >= Nn * HCOLS) return;
    int n = t >> 9, hd = (t & 511) >> 6;
    float v = num1[t] / den1[n * NHEADS + hd];
    num1[t] = v > 0.f ? v : expm1f(v);
}

// ---- layer2 per-node scores (mu: w=0, lv: w=1) ----
__global__ void scores2(const float* __restrict__ hmu, const float* __restrict__ hlv,
                        const float* __restrict__ mu_a, const float* __restrict__ lv_a,
                        float* __restrict__ s2s, float* __restrict__ s2d, int Nn) {
    int t = blockIdx.x * blockDim.x + threadIdx.x;
    if (t >= Nn * 2) return;
    int n = t >> 1, w = t & 1;
    const float* __restrict__ h = (w ? hlv : hmu) + (size_t)n * NOUT;
    const float* __restrict__ a = w ? lv_a : mu_a;
    float s0 = 0.f, s1 = 0.f;
    #pragma unroll 8
    for (int j = 0; j < NOUT; ++j) { float v = h[j]; s0 += v * a[j]; s1 += v * a[NOUT + j]; }
    s2s[w * Nn + n] = s0; s2d[w * Nn + n] = s1;
}

__global__ void edge_e2(const int* __restrict__ src, const int* __restrict__ dst,
                        const float* __restrict__ s2s, const float* __restrict__ s2d,
                        float* __restrict__ e2, float* __restrict__ den2, int E, int Nn) {
    int t = blockIdx.x * blockDim.x + threadIdx.x;
    if (t >= E * 2) return;
    int edge = t >> 1, w = t & 1;
    int si = src[edge], di = dst[edge];
    float x = s2s[w * Nn + si] + s2d[w * Nn + di];
    float l = x > 0.f ? x : ALPHA * x;
    float e = expf(-l);
    e2[t] = e;
    atomicAdd(&den2[w * Nn + si], e);
}

__global__ void scatter2(const int* __restrict__ src, const int* __restrict__ dst,
                         const float* __restrict__ e2, const float* __restrict__ hmu,
                         const float* __restrict__ hlv, float* __restrict__ nmu,
                         float* __restrict__ nlv, int E) {
    int t = blockIdx.x * blockDim.x + threadIdx.x;
    if (t >= E * 2 * NOUT) return;
    int edge = t >> 8, r = t & 255, w = r >> 7, c = r & 127;
    int si = src[edge], di = dst[edge];
    const float* __restrict__ h = w ? hlv : hmu;
    float* __restrict__ acc = w ? nlv : nmu;
    atomicAdd(&acc[(size_t)si * NOUT + c],
              e2[edge * 2 + w] * h[(size_t)di * NOUT + c]);
}

// ---- normalize mu/lv + reparameterize, write (z, mu, logvar) ----
__global__ void finalize_k(const float* __restrict__ nmu, const float* __restrict__ nlv,
                           const float* __restrict__ den2, const float* __restrict__ eps,
                           float* __restrict__ out, int Nn) {
    int t = blockIdx.x * blockDim.x + threadIdx.x;
    if (t >= Nn * NOUT) return;
    int n = t >> 7;
    float mu = nmu[t] / den2[n];
    float lv = nlv[t] / den2[Nn + n];
    float z  = eps[t] * expf(lv) + mu;
    size_t blk = (size_t)Nn * NOUT;
    out[t] = z;
    out[blk + t] = mu;
    out[2 * blk + t] = lv;
}

extern "C" void kernel_launch(void* const* d_in, const int* in_sizes, int n_in,
                              void* d_out, int out_size, void* d_ws, size_t ws_size,
                              hipStream_t stream) {
    (void)n_in; (void)out_size; (void)ws_size;
    const float* x    = (const float*)d_in[0];
    const float* Ws   = (const float*)d_in[1];
    const float* As   = (const float*)d_in[2];
    const float* mu_W = (const float*)d_in[3];
    const float* mu_a = (const float*)d_in[4];
    const float* lv_W = (const float*)d_in[5];
    const float* lv_a = (const float*)d_in[6];
    const float* eps  = (const float*)d_in[7];
    const int*   ei   = (const int*)d_in[8];

    const int Nn = in_sizes[0] / NFEAT;     // 50000
    const int E  = in_sizes[8] / 2;         // 450000
    const int* src = ei;
    const int* dst = ei + E;

    char* ws = (char*)d_ws;
    size_t off = 0;
    auto alloc = [&](size_t nf) { float* p = (float*)(ws + off);
                                  off += ((nf * sizeof(float) + 255) / 256) * 256; return p; };
    float* H    = alloc((size_t)Nn * HCOLS);
    float* Wall = alloc((size_t)NFEAT * HCOLS);
    float* s1s  = alloc((size_t)Nn * NHEADS);
    float* s1d  = alloc((size_t)Nn * NHEADS);
    float* e1   = alloc((size_t)E * NHEADS);
    float* den1 = alloc((size_t)Nn * NHEADS);
    float* num1 = alloc((size_t)Nn * HCOLS);   // becomes x2 after norm_elu
    float* hmu  = alloc((size_t)Nn * NOUT);
    float* hlv  = alloc((size_t)Nn * NOUT);
    float* s2s  = alloc((size_t)Nn * 2);
    float* s2d  = alloc((size_t)Nn * 2);
    float* e2   = alloc((size_t)E * 2);
    float* den2 = alloc((size_t)Nn * 2);
    float* nmu  = alloc((size_t)Nn * NOUT);
    float* nlv  = alloc((size_t)Nn * NOUT);

    // zero accumulators (capturable memset nodes)
    hipMemsetAsync(den1, 0, (size_t)Nn * NHEADS * sizeof(float), stream);
    hipMemsetAsync(num1, 0, (size_t)Nn * HCOLS * sizeof(float), stream);
    hipMemsetAsync(den2, 0, (size_t)Nn * 2 * sizeof(float), stream);
    hipMemsetAsync(nmu,  0, (size_t)Nn * NOUT * sizeof(float), stream);
    hipMemsetAsync(nlv,  0, (size_t)Nn * NOUT * sizeof(float), stream);

    repack_W<<<(NFEAT * HCOLS + 255) / 256, 256, 0, stream>>>(Ws, Wall);

    {   // layer-1 fused 8-head GEMM: H = x @ Wall  (50000x512 * 512x512)
        int tilesM = (Nn + 15) / 16;
        dim3 g((tilesM + 7) / 8, HCOLS / 64);
        gemm_wmma_f32<<<g, 256, 0, stream>>>(x, Wall, H, Nn, HCOLS, NFEAT);
    }
    scores1<<<(Nn * NHEADS + 255) / 256, 256, 0, stream>>>(H, As, s1s, s1d, Nn);
    edge_e1<<<(E * NHEADS + 255) / 256, 256, 0, stream>>>(src, dst, s1s, s1d, e1, den1, E);
    scatter1<<<(E * HCOLS + 255) / 256, 256, 0, stream>>>(src, dst, e1, H, num1, E);
    norm_elu<<<(Nn * HCOLS + 255) / 256, 256, 0, stream>>>(num1, den1, Nn);

    {   // layer-2 GEMMs: hmu = x2 @ mu_W, hlv = x2 @ lv_W  (50000x512 * 512x128)
        int tilesM = (Nn + 15) / 16;
        dim3 g((tilesM + 7) / 8, NOUT / 64);
        gemm_wmma_f32<<<g, 256, 0, stream>>>(num1, mu_W, hmu, Nn, NOUT, HCOLS);
        gemm_wmma_f32<<<g, 256, 0, stream>>>(num1, lv_W, hlv, Nn, NOUT, HCOLS);
    }
    scores2<<<(Nn * 2 + 255) / 256, 256, 0, stream>>>(hmu, hlv, mu_a, lv_a, s2s, s2d, Nn);
    edge_e2<<<(E * 2 + 255) / 256, 256, 0, stream>>>(src, dst, s2s, s2d, e2, den2, E, Nn);
    scatter2<<<(E * 2 * NOUT + 255) / 256, 256, 0, stream>>>(src, dst, e2, hmu, hlv, nmu, nlv, E);
    finalize_k<<<(Nn * NOUT + 255) / 256, 256, 0, stream>>>(nmu, nlv, den2, eps, (float*)d_out, Nn);
}